// SEFraudInspired_5342939316749
// MI455X (gfx1250) — compile-verified
//
#include <hip/hip_runtime.h>
#include <hip/hip_bf16.h>
#include <math.h>

#define DIN  128
#define DH   128
#define DOUT 2

typedef __attribute__((ext_vector_type(2))) float v2f;
typedef __attribute__((ext_vector_type(8))) float v8f;

__device__ __forceinline__ float sigmoidf_(float x) { return 1.0f / (1.0f + __expf(-x)); }

// Non-returning, relaxed, device-scope fp32 add -> native global_atomic_add_f32
__device__ __forceinline__ void atomAddF(float* p, float v) {
  (void)__hip_atomic_fetch_add(p, v, __ATOMIC_RELAXED, __HIP_MEMORY_SCOPE_AGENT);
}

// ---------------- degree / dinv ----------------
__global__ void k_deg_init(unsigned* __restrict__ deg, int n) {
  int i = blockIdx.x * blockDim.x + threadIdx.x;
  if (i < n) deg[i] = 1u;  // self loop contributes 1
}

__global__ void k_deg_accum(unsigned* __restrict__ deg, const int* __restrict__ dst, int e) {
  int i = blockIdx.x * blockDim.x + threadIdx.x;
  if (i < e) atomicAdd(&deg[dst[i]], 1u);  // native u32 atomic
}

// in-place: reinterpret count buffer as float rsqrt(deg)
__global__ void k_deg_to_dinv(unsigned* __restrict__ deg, int n) {
  int i = blockIdx.x * blockDim.x + threadIdx.x;
  if (i < n) {
    float d = (float)deg[i];
    float r = (d > 0.0f) ? rsqrtf(d) : 0.0f;
    ((float*)deg)[i] = r;
  }
}

// ---------------- WMMA GEMM: C[N x 128] = A[N x 128] @ W[128 x 128] ----------------
// gate != nullptr: fold sigmoid(gate[k]) into row k of W (feature-mask fusion).
// W staged in LDS pair-interleaved: float2 at [(k/2)*128 + n] = (W[k][n], W[k+1][n]).
__global__ void __launch_bounds__(256)
k_gemm128_wmma(const float* __restrict__ A, const float* __restrict__ W,
               const float* __restrict__ gate, float* __restrict__ C, int numTiles) {
  __shared__ float lw[DIN * DH];  // 64 KB of the 320 KB WGP LDS
  const int tid = threadIdx.x;

  for (int idx = tid; idx < DIN * DH; idx += 256) {
    int k = idx >> 7, nn = idx & 127;
    float w = W[idx];
    if (gate) w *= sigmoidf_(gate[k]);
    lw[((k >> 1) << 8) + (nn << 1) + (k & 1)] = w;
  }
  __syncthreads();

  const int wave = tid >> 5;         // wave32
  const int lane = tid & 31;
  const int tile = blockIdx.x * 8 + wave;
  if (tile >= numTiles) return;      // wave-uniform: EXEC stays all-ones for WMMA

  const int m     = lane & 15;       // row within tile (A) / col within tile (B,C)
  const int khalf = lane >> 4;       // 0: K={0,1}; 1: K={2,3} within each K-step of 4
  const float* arow = A + (size_t)(tile * 16 + m) * DIN + (khalf << 1);

  v8f acc[8] = {};

  #pragma unroll 4
  for (int k0 = 0; k0 < DIN; k0 += 4) {
    v2f a = *(const v2f*)(arow + k0);
    const float* bbase = &lw[(((k0 >> 1) + khalf) << 8) + (m << 1)];
    #pragma unroll
    for (int nt = 0; nt < 8; ++nt) {
      v2f b = *(const v2f*)(bbase + (nt << 5));
      acc[nt] = __builtin_amdgcn_wmma_f32_16x16x4_f32(
          false, a, false, b, (short)0, acc[nt], false, false);
    }
  }

  // D layout: VGPR v -> row (v + 8*khalf), col = lane&15
  const int row0 = tile * 16 + (khalf << 3);
  #pragma unroll
  for (int nt = 0; nt < 8; ++nt) {
    float* crow = C + (size_t)row0 * DH + nt * 16 + m;
    #pragma unroll
    for (int v = 0; v < 8; ++v) crow[(size_t)v * DH] = acc[nt][v];
  }
}

// ---------------- aggregation ----------------
// agg[i,:] = h[i,:] * dinv[i]^2   (self loop term, also initializes agg)
__global__ void k_agg_self(const float* __restrict__ h, const float* __restrict__ dinv,
                           float* __restrict__ agg, int n) {
  int g = blockIdx.x * blockDim.x + threadIdx.x;  // n*32 threads, float4 per thread
  int node = g >> 5, q = g & 31;
  if (node < n) {
    float di = dinv[node];
    float w  = di * di;
    const float4 hv = *(const float4*)(h + (size_t)node * DH + q * 4);
    float4 o = make_float4(hv.x * w, hv.y * w, hv.z * w, hv.w * w);
    *(float4*)(agg + (size_t)node * DH + q * 4) = o;
  }
}

// agg[d,:] += h[s,:] * dinv[s]*dinv[d]  per edge
__global__ void k_agg_edges(const float* __restrict__ h, const float* __restrict__ dinv,
                            const int* __restrict__ src, const int* __restrict__ dst,
                            float* __restrict__ agg, int e) {
  int g = blockIdx.x * blockDim.x + threadIdx.x;  // e*32 threads
  int ei = g >> 5, q = g & 31;
  if (ei < e) {
    int s = src[ei], d = dst[ei];
    float norm = dinv[s] * dinv[d];
    const float4 hv = *(const float4*)(h + (size_t)s * DH + q * 4);
    float* o = agg + (size_t)d * DH + q * 4;
    atomAddF(o + 0, hv.x * norm);
    atomAddF(o + 1, hv.y * norm);
    atomAddF(o + 2, hv.z * norm);
    atomAddF(o + 3, hv.w * norm);
  }
}

// x[i,:] = (x[i,:] + b)  [+ relu]
__global__ void k_bias_act(float* __restrict__ x, const float* __restrict__ b, int n, int relu) {
  int g = blockIdx.x * blockDim.x + threadIdx.x;  // n*32 threads
  int node = g >> 5, q = g & 31;
  if (node < n) {
    float4 v  = *(float4*)(x + (size_t)node * DH + q * 4);
    const float4 bv = *(const float4*)(b + q * 4);
    v.x += bv.x; v.y += bv.y; v.z += bv.z; v.w += bv.w;
    if (relu) {
      v.x = fmaxf(v.x, 0.0f); v.y = fmaxf(v.y, 0.0f);
      v.z = fmaxf(v.z, 0.0f); v.w = fmaxf(v.w, 0.0f);
    }
    *(float4*)(x + (size_t)node * DH + q * 4) = v;
  }
}

// out[i,0:2] = h[i,:] @ Wc[128x2] + bc
__global__ void k_classifier(const float* __restrict__ h, const float* __restrict__ Wc,
                             const float* __restrict__ bc, float* __restrict__ out, int n) {
  int i = blockIdx.x * blockDim.x + threadIdx.x;
  if (i < n) {
    const float* r = h + (size_t)i * DH;
    float a0 = bc[0], a1 = bc[1];
    #pragma unroll 8
    for (int k = 0; k < DH; ++k) {
      float v = r[k];
      a0 = fmaf(v, Wc[2 * k + 0], a0);
      a1 = fmaf(v, Wc[2 * k + 1], a1);
    }
    out[2 * i + 0] = a0;
    out[2 * i + 1] = a1;
  }
}

extern "C" void kernel_launch(void* const* d_in, const int* in_sizes, int n_in,
                              void* d_out, int out_size, void* d_ws, size_t ws_size,
                              hipStream_t stream) {
  const float* x    = (const float*)d_in[0];
  const int*   ei   = (const int*)  d_in[1];
  const float* mask = (const float*)d_in[2];
  const float* W1   = (const float*)d_in[3];
  const float* b1   = (const float*)d_in[4];
  const float* W2   = (const float*)d_in[5];
  const float* b2   = (const float*)d_in[6];
  const float* Wc   = (const float*)d_in[7];
  const float* bc   = (const float*)d_in[8];
  float* out = (float*)d_out;

  const int n = in_sizes[0] / DIN;   // 100000
  const int e = in_sizes[1] / 2;     // 1600000
  const int* src = ei;
  const int* dst = ei + e;

  // workspace: dinv[n] | hbuf[n*128] | agg[n*128]
  float* dinv = (float*)d_ws;
  float* hbuf = dinv + n;
  float* agg  = hbuf + (size_t)n * DH;

  const int T = 256;
  const int numTiles   = (n + 15) / 16;         // n = 100000 = 6250*16
  const int gemmBlocks = (numTiles + 7) / 8;    // 8 waves per block, 1 tile per wave
  const int nodeVec    = (n * 32 + T - 1) / T;  // float4-per-lane node passes
  const int edgeVec    = (e * 32 + T - 1) / T;

  // degree (u32 counts) -> dinv (float, in place)
  k_deg_init   <<<(n + T - 1) / T, T, 0, stream>>>((unsigned*)dinv, n);
  k_deg_accum  <<<(e + T - 1) / T, T, 0, stream>>>((unsigned*)dinv, dst, e);
  k_deg_to_dinv<<<(n + T - 1) / T, T, 0, stream>>>((unsigned*)dinv, n);

  // layer 1: (x * sigmoid(mask)) @ W1  -> aggregate -> +b1, relu
  k_gemm128_wmma<<<gemmBlocks, T, 0, stream>>>(x, W1, mask, hbuf, numTiles);
  k_agg_self    <<<nodeVec, T, 0, stream>>>(hbuf, dinv, agg, n);
  k_agg_edges   <<<edgeVec, T, 0, stream>>>(hbuf, dinv, src, dst, agg, e);
  k_bias_act    <<<nodeVec, T, 0, stream>>>(agg, b1, n, 1);

  // layer 2: h @ W2 -> aggregate -> +b2
  k_gemm128_wmma<<<gemmBlocks, T, 0, stream>>>(agg, W2, nullptr, hbuf, numTiles);
  k_agg_self    <<<nodeVec, T, 0, stream>>>(hbuf, dinv, agg, n);
  k_agg_edges   <<<edgeVec, T, 0, stream>>>(hbuf, dinv, src, dst, agg, e);
  k_bias_act    <<<nodeVec, T, 0, stream>>>(agg, b2, n, 0);

  // classifier
  k_classifier  <<<(n + T - 1) / T, T, 0, stream>>>(agg, Wc, bc, out, n);
}